// CombinedLoss_74912819576908
// MI455X (gfx1250) — compile-verified
//
#include <hip/hip_runtime.h>
#include <math.h>

typedef __attribute__((ext_vector_type(2))) float v2f;
typedef __attribute__((ext_vector_type(8))) float v8f;

#define NCLUST        40
#define DIM           24
#define HW            16384          // 128*128
#define NPIX          (16 * HW)      // 262144 pixels total
#define RECON_N       (16 * 3 * 512 * 512)
#define RECON_BLOCKS  1024
#define KMEANS_BLOCKS 2048           // 8 waves/block * 16 pixels/wave = 128 pixels/block

// ---------------------------------------------------------------------------
// Kernel 1: reconstruction MSE partial sums (pure streaming, float4 loads)
// ---------------------------------------------------------------------------
__global__ void recon_mse_partial(const float* __restrict__ x,
                                  const float* __restrict__ r,
                                  float* __restrict__ partial) {
    __shared__ float sdata[256];
    const int tid = threadIdx.x;
    size_t i = (size_t)blockIdx.x * blockDim.x + tid;
    const size_t stride = (size_t)gridDim.x * blockDim.x;
    const float4* __restrict__ x4 = (const float4*)x;
    const float4* __restrict__ r4 = (const float4*)r;
    const size_t n4 = RECON_N / 4;

    float acc = 0.f;
    for (size_t j = i; j < n4; j += stride) {
        float4 a = x4[j];
        float4 b = r4[j];
        float dx = a.x - b.x, dy = a.y - b.y, dz = a.z - b.z, dw = a.w - b.w;
        acc = fmaf(dx, dx, acc);
        acc = fmaf(dy, dy, acc);
        acc = fmaf(dz, dz, acc);
        acc = fmaf(dw, dw, acc);
    }
    sdata[tid] = acc;
    __syncthreads();
    for (int s = 128; s > 0; s >>= 1) {
        if (tid < s) sdata[tid] += sdata[tid + s];
        __syncthreads();
    }
    if (tid == 0) partial[blockIdx.x] = sdata[0];
}

// ---------------------------------------------------------------------------
// Kernel 2: soft-kmeans term via V_WMMA_F32_16X16X4_F32.
// Each wave32 handles a 16-pixel x 40-cluster tile:
//   dot(16x16 per cluster-tile) = A(16x4 f32) x B(4x16 f32) accumulated over
//   6 K-chunks (DIM=24), 3 cluster tiles (40 clusters padded to 48).
//   dist[m][n] = |e_m|^2 + |c_n|^2 - 2*dot[m][n]
// A layout (ISA 7.12.2): lanes 0-15 = {K0,K1}, lanes 16-31 = {K2,K3}, M = lane%16.
// B layout mirrors A with N = lane%16. C/D: lane = N, VGPR v = M (+8 hi half).
// ---------------------------------------------------------------------------
__global__ void kmeans_partial(const float* __restrict__ emb,
                               const float* __restrict__ cent,
                               float* __restrict__ partial) {
    __shared__ float snorm[8][32];   // per-wave per-lane partial |e|^2
    __shared__ float swsum[8];

    const int tid    = threadIdx.x;
    const int wave   = tid >> 5;
    const int lane   = tid & 31;
    const int laneLo = lane & 15;
    const bool hi    = lane >= 16;
    const int kofs   = hi ? 2 : 0;

    // 16-pixel tile for this wave; tiles never straddle an image (HW % 16 == 0)
    const int tile = blockIdx.x * 8 + wave;          // 0..16383
    const int p0   = tile * 16;
    const int nimg = p0 >> 14;                       // / 16384
    const int hw0  = p0 & (HW - 1);
    const float* __restrict__ eb = emb + (size_t)nimg * DIM * HW + hw0;

    // --- B tiles (centroids, transposed into WMMA B layout) + |c|^2 ---
    v2f   B[3][6];
    float cn[3];
    for (int ct = 0; ct < 3; ++ct) {
        const int nc = ct * 16 + laneLo;
        const bool valid = nc < NCLUST;
        float s = 0.f;
        if (valid) {
            const float* cp = cent + nc * DIM;
            #pragma unroll
            for (int d = 0; d < DIM; ++d) s = fmaf(cp[d], cp[d], s);
        }
        cn[ct] = s;
        #pragma unroll
        for (int kc = 0; kc < 6; ++kc) {
            const int k0 = kc * 4 + kofs;
            v2f b;
            b.x = valid ? cent[nc * DIM + k0]     : 0.f;
            b.y = valid ? cent[nc * DIM + k0 + 1] : 0.f;
            B[ct][kc] = b;
        }
    }

    // --- A tile (16 pixels x 24 dims), coalesced along hw, + partial |e|^2 ---
    v2f   A[6];
    float pn = 0.f;
    #pragma unroll
    for (int kc = 0; kc < 6; ++kc) {
        const int k0 = kc * 4 + kofs;
        v2f a;
        a.x = eb[(size_t)k0 * HW + laneLo];
        a.y = eb[(size_t)(k0 + 1) * HW + laneLo];
        A[kc] = a;
        pn = fmaf(a.x, a.x, pn);
        pn = fmaf(a.y, a.y, pn);
    }
    snorm[wave][lane] = pn;

    // --- 18 WMMAs: 3 cluster tiles x 6 K-chunks ---
    v8f acc0 = {}, acc1 = {}, acc2 = {};
    #pragma unroll
    for (int kc = 0; kc < 6; ++kc) {
        acc0 = __builtin_amdgcn_wmma_f32_16x16x4_f32(false, A[kc], false, B[0][kc],
                                                     (short)0, acc0, false, false);
        acc1 = __builtin_amdgcn_wmma_f32_16x16x4_f32(false, A[kc], false, B[1][kc],
                                                     (short)0, acc1, false, false);
        acc2 = __builtin_amdgcn_wmma_f32_16x16x4_f32(false, A[kc], false, B[2][kc],
                                                     (short)0, acc2, false, false);
    }
    __syncthreads();

    // --- per-pixel softmax(1/dist) weighted distance, 16-lane butterflies ---
    const bool v2ok = laneLo < (NCLUST - 32);   // clusters 32..39 only
    float lacc = 0.f;
    #pragma unroll
    for (int v = 0; v < 8; ++v) {
        const int m = v + (hi ? 8 : 0);         // pixel row this lane sees
        const float en = snorm[wave][m] + snorm[wave][m + 16];
        const float d0 = en + cn[0] - 2.f * acc0[v];
        const float d1 = en + cn[1] - 2.f * acc1[v];
        const float d2 = en + cn[2] - 2.f * acc2[v];
        const float w0 = 1.f / d0;
        const float w1 = 1.f / d1;
        const float w2 = v2ok ? 1.f / d2 : -3.0e38f;

        float wm = fmaxf(w0, fmaxf(w1, w2));
        #pragma unroll
        for (int off = 1; off < 16; off <<= 1)
            wm = fmaxf(wm, __shfl_xor(wm, off, 32));

        const float e0 = __expf(w0 - wm);
        const float e1 = __expf(w1 - wm);
        const float e2 = v2ok ? __expf(w2 - wm) : 0.f;
        float se = e0 + e1 + e2;
        float sd = fmaf(e0, d0, fmaf(e1, d1, e2 * d2));
        #pragma unroll
        for (int off = 1; off < 16; off <<= 1) {
            se += __shfl_xor(se, off, 32);
            sd += __shfl_xor(sd, off, 32);
        }
        if (laneLo == 0) {                      // lanes 0 and 16: pixels v, v+8
            const float s = sd / se;
            lacc = fmaf(s, s, lacc);
        }
    }
    lacc += __shfl_xor(lacc, 16, 32);
    if (lane == 0) swsum[wave] = lacc;
    __syncthreads();
    if (tid == 0) {
        float s = 0.f;
        #pragma unroll
        for (int w = 0; w < 8; ++w) s += swsum[w];
        partial[blockIdx.x] = s;
    }
}

// ---------------------------------------------------------------------------
// Kernel 3: combine partials into the scalar loss
// ---------------------------------------------------------------------------
__global__ void final_reduce(const float* __restrict__ ws, float* __restrict__ out) {
    __shared__ float s1[256];
    __shared__ float s2[256];
    const int tid = threadIdx.x;
    float r = 0.f, k = 0.f;
    for (int i = tid; i < RECON_BLOCKS; i += 256)  r += ws[i];
    for (int i = tid; i < KMEANS_BLOCKS; i += 256) k += ws[RECON_BLOCKS + i];
    s1[tid] = r;
    s2[tid] = k;
    __syncthreads();
    for (int s = 128; s > 0; s >>= 1) {
        if (tid < s) { s1[tid] += s1[tid + s]; s2[tid] += s2[tid + s]; }
        __syncthreads();
    }
    if (tid == 0)
        out[0] = 0.5f * (s1[0] / (float)RECON_N) + 0.5f * (s2[0] / (float)NPIX);
}

// ---------------------------------------------------------------------------
extern "C" void kernel_launch(void* const* d_in, const int* in_sizes, int n_in,
                              void* d_out, int out_size, void* d_ws, size_t ws_size,
                              hipStream_t stream) {
    const float* x     = (const float*)d_in[0];
    const float* recon = (const float*)d_in[1];
    const float* emb   = (const float*)d_in[2];
    const float* cent  = (const float*)d_in[3];
    float* out = (float*)d_out;
    float* ws  = (float*)d_ws;

    recon_mse_partial<<<RECON_BLOCKS, 256, 0, stream>>>(x, recon, ws);
    kmeans_partial<<<KMEANS_BLOCKS, 256, 0, stream>>>(emb, cent, ws + RECON_BLOCKS);
    final_reduce<<<1, 256, 0, stream>>>(ws, out);
}